// GraphSageModel_61349312856089
// MI455X (gfx1250) — compile-verified
//
#include <hip/hip_runtime.h>
#include <hip/hip_bf16.h>

#define N_NODES_C 100000
#define N_EDGES_C 1600000
#define IN_DIM_C  64
#define H0_C      128
#define H1_C      128
#define H2_C      256
#define OUT_DIM_C 40

typedef __attribute__((ext_vector_type(2))) float v2f;
typedef __attribute__((ext_vector_type(8))) float v8f;

// ---------------------------------------------------------------------------
// Utility: grid-stride zero
// ---------------------------------------------------------------------------
__global__ void zero_f32(float* __restrict__ p, long long n) {
    long long i = (long long)blockIdx.x * blockDim.x + threadIdx.x;
    long long stride = (long long)gridDim.x * blockDim.x;
    for (; i < n; i += stride) p[i] = 0.0f;
}

// ---------------------------------------------------------------------------
// Edge degree count (f32 so the divide matches the reference exactly)
// ---------------------------------------------------------------------------
__global__ void edge_count(const long long* __restrict__ dst,
                           float* __restrict__ cnt, int nE) {
    int e = blockIdx.x * blockDim.x + threadIdx.x;
    if (e < nE) {
        __hip_atomic_fetch_add(&cnt[dst[e]], 1.0f,
                               __ATOMIC_RELAXED, __HIP_MEMORY_SCOPE_AGENT);
    }
}

// ---------------------------------------------------------------------------
// Scatter-add of source features into destination accumulator.
// One thread handles a float4 chunk of one edge: vector gather + 4 HW f32
// atomics (no return -> STOREcnt path, resolved in L2).
// ---------------------------------------------------------------------------
template <int D>
__global__ void edge_scatter(const float* __restrict__ X,
                             const long long* __restrict__ src,
                             const long long* __restrict__ dst,
                             float* __restrict__ agg, int nE) {
    const int CH = D / 4;
    int tid = blockIdx.x * blockDim.x + threadIdx.x;
    int e = tid / CH;
    int c = tid - e * CH;
    if (e >= nE) return;
    long long s = src[e];
    long long d = dst[e];
    const float4 v = reinterpret_cast<const float4*>(X + s * D)[c];
    float* o = agg + d * D + (long long)c * 4;
    __hip_atomic_fetch_add(o + 0, v.x, __ATOMIC_RELAXED, __HIP_MEMORY_SCOPE_AGENT);
    __hip_atomic_fetch_add(o + 1, v.y, __ATOMIC_RELAXED, __HIP_MEMORY_SCOPE_AGENT);
    __hip_atomic_fetch_add(o + 2, v.z, __ATOMIC_RELAXED, __HIP_MEMORY_SCOPE_AGENT);
    __hip_atomic_fetch_add(o + 3, v.w, __ATOMIC_RELAXED, __HIP_MEMORY_SCOPE_AGENT);
}

// ---------------------------------------------------------------------------
// Fused GraphSAGE layer: OUT = act( [X | AGG/max(CNT,1)] @ W + b )
// Exact fp32 via V_WMMA_F32_16X16X4_F32. One wave per 16x16 output tile.
// A-frag (16x4):  lane&15 = row, lane>=16 holds K+2/K+3 (ISA 7.12.2)
// B-frag (4x16):  lane&15 = col, lane>=16 holds K+2/K+3
// C/D (16x16):    VGPR v -> row v (lanes 0-15) / row v+8 (lanes 16-31)
// ---------------------------------------------------------------------------
template <int D, int H, int ACT>   // ACT: 0 = none, 1 = sigmoid
__global__ void sage_gemm(const float* __restrict__ X,
                          const float* __restrict__ AGG,
                          const float* __restrict__ CNT,
                          const float* __restrict__ W,    // (2D) x H row-major
                          const float* __restrict__ Bs,   // H
                          float* __restrict__ OUT,        // N x H
                          int totalWaves) {
    int w = blockIdx.x * (blockDim.x >> 5) + (threadIdx.x >> 5);
    if (w >= totalWaves) return;
    const int colTiles = H / 16;
    int rowTile = w / colTiles;
    int colTile = w - rowTile * colTiles;
    int lane = threadIdx.x & 31;

    int mRow = rowTile * 16 + (lane & 15);       // A row this lane loads
    int kOff = (lane >> 4) * 2;                  // 0 for lanes 0-15, 2 for 16-31
    int nCol = colTile * 16 + (lane & 15);       // B/D column this lane owns

    const float invc = 1.0f / fmaxf(CNT[mRow], 1.0f);
    const float* xrow = X + (long long)mRow * D;
    const float* arow = AGG + (long long)mRow * D;

    v8f c = {};
    // self half: k in [0, D)
    #pragma unroll 4
    for (int k = 0; k < D; k += 4) {
        v2f a, b;
        a.x = xrow[k + kOff];
        a.y = xrow[k + kOff + 1];
        int kb = k + kOff;
        b.x = W[(long long)kb * H + nCol];
        b.y = W[(long long)(kb + 1) * H + nCol];
        c = __builtin_amdgcn_wmma_f32_16x16x4_f32(false, a, false, b,
                                                  (short)0, c, false, false);
    }
    // neighbor-mean half: k in [D, 2D)
    #pragma unroll 4
    for (int k = 0; k < D; k += 4) {
        v2f a, b;
        a.x = arow[k + kOff] * invc;
        a.y = arow[k + kOff + 1] * invc;
        int kb = D + k + kOff;
        b.x = W[(long long)kb * H + nCol];
        b.y = W[(long long)(kb + 1) * H + nCol];
        c = __builtin_amdgcn_wmma_f32_16x16x4_f32(false, a, false, b,
                                                  (short)0, c, false, false);
    }

    const float bias = Bs[nCol];
    int mBase = rowTile * 16 + ((lane < 16) ? 0 : 8);
    #pragma unroll
    for (int v = 0; v < 8; ++v) {
        float val = c[v] + bias;
        if (ACT == 1) val = 1.0f / (1.0f + __expf(-val));
        OUT[(long long)(mBase + v) * H + nCol] = val;
    }
}

// ---------------------------------------------------------------------------
// Plain GEMM + bias + activation (handles H not multiple of 16 via masking).
// ---------------------------------------------------------------------------
template <int K, int H, int ACT>   // ACT: 0 = none, 2 = relu
__global__ void mlp_gemm(const float* __restrict__ X,   // N x K
                         const float* __restrict__ W,   // K x H row-major
                         const float* __restrict__ Bs,  // H
                         float* __restrict__ OUT,       // N x H
                         int totalWaves) {
    int w = blockIdx.x * (blockDim.x >> 5) + (threadIdx.x >> 5);
    if (w >= totalWaves) return;
    const int colTiles = (H + 15) / 16;
    int rowTile = w / colTiles;
    int colTile = w - rowTile * colTiles;
    int lane = threadIdx.x & 31;

    int mRow = rowTile * 16 + (lane & 15);
    int kOff = (lane >> 4) * 2;
    int nCol = colTile * 16 + (lane & 15);
    const bool colOK = (nCol < H);

    const float* xrow = X + (long long)mRow * K;

    v8f c = {};
    #pragma unroll 4
    for (int k = 0; k < K; k += 4) {
        v2f a, b;
        a.x = xrow[k + kOff];
        a.y = xrow[k + kOff + 1];
        int kb = k + kOff;
        b.x = colOK ? W[(long long)kb * H + nCol] : 0.0f;
        b.y = colOK ? W[(long long)(kb + 1) * H + nCol] : 0.0f;
        c = __builtin_amdgcn_wmma_f32_16x16x4_f32(false, a, false, b,
                                                  (short)0, c, false, false);
    }

    const float bias = colOK ? Bs[nCol] : 0.0f;
    int mBase = rowTile * 16 + ((lane < 16) ? 0 : 8);
    if (colOK) {
        #pragma unroll
        for (int v = 0; v < 8; ++v) {
            float val = c[v] + bias;
            if (ACT == 2) val = fmaxf(val, 0.0f);
            OUT[(long long)(mBase + v) * H + nCol] = val;
        }
    }
}

// ---------------------------------------------------------------------------
// Host-side orchestration
// ---------------------------------------------------------------------------
extern "C" void kernel_launch(void* const* d_in, const int* in_sizes, int n_in,
                              void* d_out, int out_size, void* d_ws, size_t ws_size,
                              hipStream_t stream) {
    const float*     features = (const float*)d_in[0];
    const long long* eidx     = (const long long*)d_in[1];   // int64 per reference
    const float*     W1  = (const float*)d_in[2];
    const float*     b1  = (const float*)d_in[3];
    const float*     W2  = (const float*)d_in[4];
    const float*     b2  = (const float*)d_in[5];
    const float*     Wm1 = (const float*)d_in[6];
    const float*     bm1 = (const float*)d_in[7];
    const float*     Wm2 = (const float*)d_in[8];
    const float*     bm2 = (const float*)d_in[9];
    float* out = (float*)d_out;

    const long long N = N_NODES_C;
    const long long E = N_EDGES_C;
    const long long* src = eidx;        // edge_index[0]
    const long long* dst = eidx + E;    // edge_index[1]

    // Workspace layout (floats): cnt(N) | agg(N*128) | h1(N*128) | h2(N*128)
    // h3 (N*256) aliases agg+h1 once both are dead. Total = N*385 f32 ~154 MB.
    float* ws  = (float*)d_ws;
    float* cnt = ws;
    float* agg = ws + N;
    float* h1  = agg + N * 128;
    float* h2  = h1 + N * 128;
    float* h3  = agg;                   // spans agg + h1 regions (contiguous)

    // Degree counts (edge-structure only: compute once).
    zero_f32<<<1024, 256, 0, stream>>>(cnt, N);
    edge_count<<<(int)((E + 255) / 256), 256, 0, stream>>>(dst, cnt, (int)E);

    // ---- SAGE layer 1 (D=64 -> H=128, sigmoid) ----
    zero_f32<<<4096, 256, 0, stream>>>(agg, N * IN_DIM_C);
    {
        long long work = E * (IN_DIM_C / 4);
        edge_scatter<IN_DIM_C><<<(int)((work + 255) / 256), 256, 0, stream>>>(
            features, src, dst, agg, (int)E);
    }
    {
        int waves = (int)(N / 16) * (H0_C / 16);          // 6250 * 8 = 50000
        sage_gemm<IN_DIM_C, H0_C, 1><<<(waves + 3) / 4, 128, 0, stream>>>(
            features, agg, cnt, W1, b1, h1, waves);
    }

    // ---- SAGE layer 2 (D=128 -> H=128, sigmoid) ----
    zero_f32<<<8192, 256, 0, stream>>>(agg, N * H0_C);
    {
        long long work = E * (H0_C / 4);
        edge_scatter<H0_C><<<(int)((work + 255) / 256), 256, 0, stream>>>(
            h1, src, dst, agg, (int)E);
    }
    {
        int waves = (int)(N / 16) * (H1_C / 16);          // 50000
        sage_gemm<H0_C, H1_C, 1><<<(waves + 3) / 4, 128, 0, stream>>>(
            h1, agg, cnt, W2, b2, h2, waves);
    }

    // ---- MLP hidden: h3 = relu(h2 @ Wm1 + bm1)  (agg/h1 now dead -> h3) ----
    {
        int waves = (int)(N / 16) * (H2_C / 16);          // 6250 * 16 = 100000
        mlp_gemm<H1_C, H2_C, 2><<<(waves + 3) / 4, 128, 0, stream>>>(
            h2, Wm1, bm1, h3, waves);
    }

    // ---- MLP out: out = h3 @ Wm2 + bm2 (40 cols -> 3 masked tiles) ----
    {
        int waves = (int)(N / 16) * ((OUT_DIM_C + 15) / 16);  // 6250 * 3 = 18750
        mlp_gemm<H2_C, OUT_DIM_C, 0><<<(waves + 3) / 4, 128, 0, stream>>>(
            h3, Wm2, bm2, out, waves);
    }
}